// LinearCSRForward_51299089383831
// MI455X (gfx1250) — compile-verified
//
#include <hip/hip_runtime.h>

typedef __attribute__((ext_vector_type(2))) float v2f;
typedef __attribute__((ext_vector_type(8))) float v8f;

#define M_DIM 32
#define N_DIM 11008
#define K_DIM 4096

// Each wave computes a 32(M) x 32(N) output tile for one K-segment:
//   4 accumulators (16x16 each), 2 A-fragments x 2 B-fragments per k-step.
// A-fragment (16x4 f32): lane<16 -> rows 0..15 holding k,k+1 ; lane>=16 holds k+2,k+3.
// B-fragment (4x16 f32): lane<16 -> col N=lane holding k,k+1 ; lane>=16 holds k+2,k+3.
// C/D (16x16 f32): vgpr r, laneLo=N, laneHi selects M=r or M=8+r.
__global__ __launch_bounds__(128) void gemm_xwt_wmma_f32(
    const float* __restrict__ x,     // 32 x 4096
    const float* __restrict__ W,     // 11008 x 4096 (row-major; we need W^T)
    const float* __restrict__ bias,  // 11008
    float* __restrict__ dest,        // d_out (kseg==1) or partials ws
    int klen, int addBias)
{
    const int lane   = threadIdx.x & 31;
    const int wave   = threadIdx.x >> 5;     // 4 waves / block
    const int laneLo = lane & 15;
    const int laneHi = lane >> 4;

    const int n0     = (blockIdx.x * 4 + wave) * 32;   // N-tile base (32 wide)
    const int seg    = blockIdx.y;                     // K segment
    const int kStart = seg * klen;

    const float* aPtr0 = x + (size_t)laneLo * K_DIM + kStart + 2 * laneHi;      // rows 0..15
    const float* aPtr1 = aPtr0 + (size_t)16 * K_DIM;                            // rows 16..31
    const float* bPtr0 = W + (size_t)(n0 + laneLo) * K_DIM + kStart + 2 * laneHi; // cols n0..n0+15
    const float* bPtr1 = bPtr0 + (size_t)16 * K_DIM;                            // cols n0+16..n0+31

    v8f c00 = {};  // rows 0..15 , cols n0..n0+15
    v8f c01 = {};  // rows 0..15 , cols n0+16..n0+31
    v8f c10 = {};  // rows 16..31, cols n0..n0+15
    v8f c11 = {};  // rows 16..31, cols n0+16..n0+31

    // Outer iteration covers 32 K-values (8 WMMA k-steps); klen is a multiple of 32.
    for (int kk = 0; kk < klen; kk += 32) {
        // Stream-prefetch the W rows ~2KB ahead (speculative; OOB drops silently).
        __builtin_prefetch(bPtr0 + 512, 0, 0);
        __builtin_prefetch(bPtr1 + 512, 0, 0);
#pragma unroll
        for (int u = 0; u < 8; ++u) {
            v2f b0 = *(const v2f*)(bPtr0 + 4 * u);
            v2f b1 = *(const v2f*)(bPtr1 + 4 * u);
            v2f a0 = *(const v2f*)(aPtr0 + 4 * u);
            v2f a1 = *(const v2f*)(aPtr1 + 4 * u);
            // (neg_a, A, neg_b, B, c_mod, C, reuse_a, reuse_b)
            c00 = __builtin_amdgcn_wmma_f32_16x16x4_f32(false, a0, false, b0,
                                                        (short)0, c00, false, false);
            c01 = __builtin_amdgcn_wmma_f32_16x16x4_f32(false, a0, false, b1,
                                                        (short)0, c01, false, false);
            c10 = __builtin_amdgcn_wmma_f32_16x16x4_f32(false, a1, false, b0,
                                                        (short)0, c10, false, false);
            c11 = __builtin_amdgcn_wmma_f32_16x16x4_f32(false, a1, false, b1,
                                                        (short)0, c11, false, false);
        }
        aPtr0 += 32; aPtr1 += 32; bPtr0 += 32; bPtr1 += 32;
    }

    const float bv0 = addBias ? bias[n0 + laneLo]      : 0.0f;
    const float bv1 = addBias ? bias[n0 + 16 + laneLo] : 0.0f;
    float* dst = dest + (size_t)seg * M_DIM * N_DIM;
#pragma unroll
    for (int r = 0; r < 8; ++r) {
        const int m0 = r + 8 * laneHi;       // rows 0..15 block
        const int m1 = 16 + m0;              // rows 16..31 block
        dst[(size_t)m0 * N_DIM + (n0 + laneLo)]      = c00[r] + bv0;
        dst[(size_t)m0 * N_DIM + (n0 + 16 + laneLo)] = c01[r] + bv1;
        dst[(size_t)m1 * N_DIM + (n0 + laneLo)]      = c10[r] + bv0;
        dst[(size_t)m1 * N_DIM + (n0 + 16 + laneLo)] = c11[r] + bv1;
    }
}

// out[m][n] = bias[n] + sum_seg ws[seg][m][n]
__global__ __launch_bounds__(256) void reduce_bias_kernel(
    const float* __restrict__ ws,
    const float* __restrict__ bias,
    float* __restrict__ out, int kseg)
{
    const int total = M_DIM * N_DIM;
    int idx = blockIdx.x * blockDim.x + threadIdx.x;
    if (idx >= total) return;
    int n = idx % N_DIM;
    float acc = bias[n];
    for (int s = 0; s < kseg; ++s)
        acc += ws[(size_t)s * total + idx];
    out[idx] = acc;
}

extern "C" void kernel_launch(void* const* d_in, const int* in_sizes, int n_in,
                              void* d_out, int out_size, void* d_ws, size_t ws_size,
                              hipStream_t stream) {
    const float* x    = (const float*)d_in[0];
    const float* W    = (const float*)d_in[1];
    const float* bias = (const float*)d_in[2];
    float* out = (float*)d_out;

    // Pick the deepest K-split that fits in workspace (more waves -> hide HBM latency).
    const size_t per_seg = (size_t)M_DIM * N_DIM * sizeof(float);
    int kseg = 1;
    if      (ws_size >= 16 * per_seg) kseg = 16;
    else if (ws_size >=  8 * per_seg) kseg = 8;
    else if (ws_size >=  4 * per_seg) kseg = 4;
    else if (ws_size >=  2 * per_seg) kseg = 2;
    const int klen = K_DIM / kseg;   // 256..4096, all multiples of 32

    dim3 grid(N_DIM / (32 * 4), kseg);   // 86 x kseg blocks, 4 waves each
    if (kseg == 1) {
        gemm_xwt_wmma_f32<<<grid, 128, 0, stream>>>(x, W, bias, out, klen, 1);
    } else {
        float* ws = (float*)d_ws;
        gemm_xwt_wmma_f32<<<grid, 128, 0, stream>>>(x, W, bias, ws, klen, 0);
        const int total = M_DIM * N_DIM;
        reduce_bias_kernel<<<(total + 255) / 256, 256, 0, stream>>>(ws, bias, out, kseg);
    }
}